// GCN_60344290508978
// MI455X (gfx1250) — compile-verified
//
#include <hip/hip_runtime.h>

#define N_NODES 100000
#define N_EDGES 1600000

typedef float v2f __attribute__((ext_vector_type(2)));
typedef float v8f __attribute__((ext_vector_type(8)));

// ---------------- degree / normalization ----------------

__global__ void deg_init_kernel(float* __restrict__ deg, int n) {
  int i = blockIdx.x * blockDim.x + threadIdx.x;
  if (i < n) deg[i] = 1.0f;                 // self-loop contributes 1
}

__global__ void deg_accum_kernel(const long long* __restrict__ dst,
                                 float* __restrict__ deg, int e) {
  int i = blockIdx.x * blockDim.x + threadIdx.x;
  if (i < e) atomicAdd(&deg[(int)dst[i]], 1.0f);
}

__global__ void rsqrt_kernel(float* __restrict__ d, int n) {
  int i = blockIdx.x * blockDim.x + threadIdx.x;
  if (i < n) d[i] = rsqrtf(d[i]);           // deg -> dinv in place
}

__global__ void edgew_kernel(const long long* __restrict__ src,
                             const long long* __restrict__ dst,
                             const float* __restrict__ dinv,
                             float* __restrict__ w, int e) {
  int i = blockIdx.x * blockDim.x + threadIdx.x;
  if (i < e) w[i] = dinv[(int)src[i]] * dinv[(int)dst[i]];
}

__global__ void zero_kernel(float* __restrict__ p, int n) {
  int i = blockIdx.x * blockDim.x + threadIdx.x;
  if (i < n) p[i] = 0.0f;
}

// ---------------- dense projection via V_WMMA_F32_16X16X4_F32 ----------------
// One wave computes one 16x16 output tile of Out = A[N,FIN] * W[FIN,FOUT].
// fp32 A/B layout (ISA 7.12.2): lane%16 = M (for A) / N (for B,D); lane/16
// selects K pair {0,1} vs {2,3} within each K=4 step.

template <int FIN, int FOUT>
__global__ void gemm_wmma_kernel(const float* __restrict__ A,
                                 const float* __restrict__ W,
                                 const float* __restrict__ bias,  // may be null
                                 float* __restrict__ Out,
                                 int nRowTiles) {
  constexpr int COLT = (FOUT + 15) / 16;
  int wave = blockIdx.x * (blockDim.x >> 5) + (threadIdx.x >> 5);
  int rowTile = wave / COLT;
  int colTile = wave - rowTile * COLT;
  if (rowTile >= nRowTiles) return;         // wave-uniform: EXEC stays all-1s
  int lane = threadIdx.x & 31;
  int idx  = lane & 15;
  int half = lane >> 4;                     // 0: K={0,1}, 1: K={2,3}
  int row  = rowTile * 16 + idx;            // A-matrix row for this lane
  int ncol = colTile * 16 + idx;            // B/D column for this lane
  const bool colOk = (FOUT >= 16) || (ncol < FOUT);

  v8f acc = {};
#pragma unroll
  for (int k0 = 0; k0 < FIN; k0 += 4) {
    int ka = k0 + 2 * half;
    v2f a = *(const v2f*)(A + row * FIN + ka);   // 8B aligned (ka even)
    v2f b;
    b.x = colOk ? W[ka * FOUT + ncol]       : 0.0f;
    b.y = colOk ? W[(ka + 1) * FOUT + ncol] : 0.0f;
    acc = __builtin_amdgcn_wmma_f32_16x16x4_f32(
        /*neg_a=*/false, a, /*neg_b=*/false, b,
        /*c_mod=*/(short)0, acc, /*reuse_a=*/false, /*reuse_b=*/false);
  }

  if (colOk) {
    float bv = bias ? bias[ncol] : 0.0f;
#pragma unroll
    for (int r = 0; r < 8; ++r) {
      int mrow = rowTile * 16 + r + 8 * half;
      Out[mrow * FOUT + ncol] = acc[r] + bv;
    }
  }
}

// ---------------- sparse aggregation (edge-parallel, L2-resident atomics) ----

template <int F>
__global__ void aggregate_kernel(const float* __restrict__ P,
                                 const long long* __restrict__ src,
                                 const long long* __restrict__ dst,
                                 const float* __restrict__ w,
                                 float* __restrict__ agg, int e) {
  int t = blockIdx.x * blockDim.x + threadIdx.x;
  int edge = t / F;
  int f = t - edge * F;
  if (edge >= e) return;
  int s = (int)src[edge];
  int d = (int)dst[edge];
  atomicAdd(&agg[d * F + f], P[s * F + f] * w[edge]);
}

// self-loop + bias + activation epilogue
template <int F, bool TANH>
__global__ void finish_kernel(const float* __restrict__ agg,
                              const float* __restrict__ P,
                              const float* __restrict__ dinv,
                              const float* __restrict__ bias,
                              float* __restrict__ out, int n) {
  int t = blockIdx.x * blockDim.x + threadIdx.x;
  if (t >= n) return;
  int i = t / F;
  int f = t - i * F;
  float di = dinv[i];
  float v = agg[t] + P[t] * (di * di) + bias[f];
  out[t] = TANH ? tanhf(v) : v;
}

// ---------------- launcher ----------------

static inline int cdiv(long long a, long long b) { return (int)((a + b - 1) / b); }

extern "C" void kernel_launch(void* const* d_in, const int* in_sizes, int n_in,
                              void* d_out, int out_size, void* d_ws, size_t ws_size,
                              hipStream_t stream) {
  (void)in_sizes; (void)n_in; (void)out_size; (void)ws_size;
  const int N = N_NODES, E = N_EDGES;

  const float*     x   = (const float*)d_in[0];
  const long long* ei  = (const long long*)d_in[1];   // int64 [2,E]
  const long long* src = ei;
  const long long* dst = ei + E;
  const float* W1 = (const float*)d_in[2]; const float* b1 = (const float*)d_in[3];
  const float* W2 = (const float*)d_in[4]; const float* b2 = (const float*)d_in[5];
  const float* W3 = (const float*)d_in[6]; const float* b3 = (const float*)d_in[7];
  const float* Wc = (const float*)d_in[8]; const float* bc = (const float*)d_in[9];

  float* out  = (float*)d_out;            // [N,8] logits then [N,16] h3
  float* hout = out + (size_t)N * 8;

  float* ws   = (float*)d_ws;
  float* dinv = ws;                       // N floats (deg -> dinv in place)
  float* P    = dinv + N;                 // N*32 projection buffer
  float* Agg  = P + (size_t)N * 32;       // N*32 aggregation / h buffer
  float* wE   = Agg + (size_t)N * 32;     // E per-edge norm weights

  const int T = 256;
  const int rowTiles = N / 16;            // 6250, exact

  // normalization: deg (with self-loop) -> dinv -> per-edge weights
  deg_init_kernel<<<cdiv(N, T), T, 0, stream>>>(dinv, N);
  deg_accum_kernel<<<cdiv(E, T), T, 0, stream>>>(dst, dinv, E);
  rsqrt_kernel<<<cdiv(N, T), T, 0, stream>>>(dinv, N);
  edgew_kernel<<<cdiv(E, T), T, 0, stream>>>(src, dst, dinv, wE, E);

  // layer 1: 32 -> 32
  gemm_wmma_kernel<32, 32><<<cdiv(rowTiles * 2, 8), T, 0, stream>>>(x, W1, nullptr, P, rowTiles);
  zero_kernel<<<cdiv((long long)N * 32, T), T, 0, stream>>>(Agg, N * 32);
  aggregate_kernel<32><<<cdiv((long long)E * 32, T), T, 0, stream>>>(P, src, dst, wE, Agg, E);
  finish_kernel<32, true><<<cdiv((long long)N * 32, T), T, 0, stream>>>(Agg, P, dinv, b1, Agg, N * 32);

  // layer 2: 32 -> 32
  gemm_wmma_kernel<32, 32><<<cdiv(rowTiles * 2, 8), T, 0, stream>>>(Agg, W2, nullptr, P, rowTiles);
  zero_kernel<<<cdiv((long long)N * 32, T), T, 0, stream>>>(Agg, N * 32);
  aggregate_kernel<32><<<cdiv((long long)E * 32, T), T, 0, stream>>>(P, src, dst, wE, Agg, E);
  finish_kernel<32, true><<<cdiv((long long)N * 32, T), T, 0, stream>>>(Agg, P, dinv, b2, Agg, N * 32);

  // layer 3: 32 -> 16 ; h3 written straight into output tuple slot
  gemm_wmma_kernel<32, 16><<<cdiv(rowTiles * 1, 8), T, 0, stream>>>(Agg, W3, nullptr, P, rowTiles);
  zero_kernel<<<cdiv((long long)N * 16, T), T, 0, stream>>>(Agg, N * 16);
  aggregate_kernel<16><<<cdiv((long long)E * 16, T), T, 0, stream>>>(P, src, dst, wE, Agg, E);
  finish_kernel<16, true><<<cdiv((long long)N * 16, T), T, 0, stream>>>(Agg, P, dinv, b3, hout, N * 16);

  // classifier: 16 -> 8, bias fused, no activation
  gemm_wmma_kernel<16, 8><<<cdiv(rowTiles * 1, 8), T, 0, stream>>>(hout, Wc, bc, out, rowTiles);
}